// EfficientNL_7730941132980
// MI455X (gfx1250) — compile-verified
//
#include <hip/hip_runtime.h>
#include <math.h>

typedef __attribute__((ext_vector_type(2))) float v2f;
typedef __attribute__((ext_vector_type(8))) float v8f;

typedef int v4i_gcc __attribute__((vector_size(16)));
typedef __attribute__((address_space(1))) v4i_gcc glob_v4i;
typedef __attribute__((address_space(3))) v4i_gcc lds_v4i;

#define NBATCH 2
#define CIN 64
#define SLEN 262144   /* 16*128*128 */
#define KCH 32
#define VCH 32
#define RECSZ 576     /* per-block record: kmax[32] esum[32] ctx[512] */

// D(16x16,f32) = A(16x4,f32) x B(4x16,f32) + C
__device__ __forceinline__ v8f wmma4(v2f a, v2f b, v8f c) {
  return __builtin_amdgcn_wmma_f32_16x16x4_f32(false, a, false, b, (short)0, c,
                                               false, false);
}

__device__ __forceinline__ v8f vzero8() {
  v8f z;
#pragma unroll
  for (int i = 0; i < 8; ++i) z[i] = 0.0f;
  return z;
}

// ============ Pass 1 (fused): block kmax + e/v + block ctx partials =========
// Each block: 512 positions. Loop A computes per-block channel max of Wk@x
// (tile stays hot in L2 for loop B -> only one HBM pass over x here).
// Loop B: e=exp(k+bk-m_b), v=Wv@x+bv, ctx_b += E V^T (WMMA from LDS), esum_b.
__global__ void __launch_bounds__(256, 2)
k_stats(const float* __restrict__ x,
        const float* __restrict__ Wk, const float* __restrict__ bk,
        const float* __restrict__ Wv, const float* __restrict__ bv,
        float* __restrict__ partials) {
  __shared__ float eS[32][130];
  __shared__ float vS[32][130];
  __shared__ float lmax[8][32];
  __shared__ float kmaxS[32];
  __shared__ float cbuf[8][256];
  __shared__ float esbuf[8][32];

  const int tid = threadIdx.x;
  const int lane = tid & 31, wv = tid >> 5;
  const int lm = lane & 15, lh = lane >> 4;
  const int nb = blockIdx.y;
  const size_t sbase = (size_t)blockIdx.x * 512;
  const float* xb = x + (size_t)nb * CIN * SLEN;

  // ---------- Loop A: per-block channel max of k = Wk@x ----------
  v8f mx0, mx1;
#pragma unroll
  for (int r = 0; r < 8; ++r) { mx0[r] = -3.402823e38f; mx1[r] = -3.402823e38f; }

  for (int c = 0; c < 4; ++c) {
    const size_t s0 = sbase + (size_t)c * 128 + (size_t)wv * 16;
    v2f barr[16];
#pragma unroll
    for (int kk = 0; kk < 16; ++kk) {   // batched loads -> one wait
      const int row = kk * 4 + 2 * lh;
      barr[kk].x = xb[(size_t)row * SLEN + s0 + lm];
      barr[kk].y = xb[(size_t)(row + 1) * SLEN + s0 + lm];
    }
    v8f a0 = vzero8(), a1 = vzero8();
#pragma unroll
    for (int kk = 0; kk < 16; ++kk) {
      const int row = kk * 4 + 2 * lh;
      v2f w0, w1;
      const float* p;
      p = Wk + lm * CIN + row;        w0.x = p[0]; w0.y = p[1];
      p = Wk + (16 + lm) * CIN + row; w1.x = p[0]; w1.y = p[1];
      a0 = wmma4(w0, barr[kk], a0);
      a1 = wmma4(w1, barr[kk], a1);
    }
#pragma unroll
    for (int r = 0; r < 8; ++r) {
      mx0[r] = fmaxf(mx0[r], a0[r]);
      mx1[r] = fmaxf(mx1[r], a1[r]);
    }
  }
#pragma unroll
  for (int r = 0; r < 8; ++r) {
    mx0[r] += bk[r + 8 * lh];
    mx1[r] += bk[16 + r + 8 * lh];
  }
  // butterfly max within each 16-lane group (wave32)
#pragma unroll
  for (int off = 1; off < 16; off <<= 1) {
#pragma unroll
    for (int r = 0; r < 8; ++r) {
      mx0[r] = fmaxf(mx0[r], __shfl_xor(mx0[r], off, 32));
      mx1[r] = fmaxf(mx1[r], __shfl_xor(mx1[r], off, 32));
    }
  }
  if (lm == 0) {
#pragma unroll
    for (int r = 0; r < 8; ++r) {
      lmax[wv][8 * lh + r]      = mx0[r];
      lmax[wv][16 + 8 * lh + r] = mx1[r];
    }
  }
  __syncthreads();
  if (tid < 32) {
    float m = lmax[0][tid];
    for (int w = 1; w < 8; ++w) m = fmaxf(m, lmax[w][tid]);
    kmaxS[tid] = m;
  }
  __syncthreads();

  // Opaque copies: block LICM/CSE from keeping loop-B weight fragments alive
  // across loop A (this is what caused scratch spills in the MMA chain).
  const float* WkB = Wk;
  const float* WvB = Wv;
  asm volatile("" : "+s"(WkB), "+s"(WvB));

  // fused per-lane constants: bkm = bk - kmax (folded), bv
  float bkm0[8], bkm1[8], bv0[8], bv1[8];
#pragma unroll
  for (int r = 0; r < 8; ++r) {
    const int c0 = r + 8 * lh, c1 = 16 + r + 8 * lh;
    bkm0[r] = bk[c0] - kmaxS[c0];
    bkm1[r] = bk[c1] - kmaxS[c1];
    bv0[r] = bv[c0]; bv1[r] = bv[c1];
  }

  // ---------- Loop B: e, v, ctx partial, esum partial ----------
  const int head = wv & 1;
  const int quarter = wv >> 1;
  v8f ctxacc = vzero8();
  float esum = 0.0f;
  const int ech = tid & 31, egrp = tid >> 5;

  for (int c = 0; c < 4; ++c) {
    const size_t s0 = sbase + (size_t)c * 128 + (size_t)wv * 16;
    v2f barr[16];
#pragma unroll
    for (int kk = 0; kk < 16; ++kk) {   // L2-hot re-read, batched
      const int row = kk * 4 + 2 * lh;
      barr[kk].x = xb[(size_t)row * SLEN + s0 + lm];
      barr[kk].y = xb[(size_t)(row + 1) * SLEN + s0 + lm];
    }
    v8f k0 = vzero8(), k1 = vzero8(), v0 = vzero8(), v1 = vzero8();
#pragma unroll
    for (int kk = 0; kk < 16; ++kk) {
      const int row = kk * 4 + 2 * lh;
      v2f ak0, ak1, av0, av1;
      const float* p;
      p = WkB + lm * CIN + row;        ak0.x = p[0]; ak0.y = p[1];
      p = WkB + (16 + lm) * CIN + row; ak1.x = p[0]; ak1.y = p[1];
      p = WvB + lm * CIN + row;        av0.x = p[0]; av0.y = p[1];
      p = WvB + (16 + lm) * CIN + row; av1.x = p[0]; av1.y = p[1];
      k0 = wmma4(ak0, barr[kk], k0);
      k1 = wmma4(ak1, barr[kk], k1);
      v0 = wmma4(av0, barr[kk], v0);
      v1 = wmma4(av1, barr[kk], v1);
    }
    const int col = wv * 16 + lm;
#pragma unroll
    for (int r = 0; r < 8; ++r) {
      const int c0 = r + 8 * lh, c1 = 16 + r + 8 * lh;
      eS[c0][col] = __expf(k0[r] + bkm0[r]);
      eS[c1][col] = __expf(k1[r] + bkm1[r]);
      vS[c0][col] = v0[r] + bv0[r];
      vS[c1][col] = v1[r] + bv1[r];
    }
    __syncthreads();
    // ctx += E_head(16 x 4pos) x Vt(4pos x 16) via WMMA, fragments from LDS
#pragma unroll
    for (int kp = 0; kp < 8; ++kp) {
      const int pos = quarter * 32 + kp * 4 + 2 * lh;
      v2f a, b2;
      a.x  = eS[head * 16 + lm][pos];
      a.y  = eS[head * 16 + lm][pos + 1];
      b2.x = vS[head * 16 + lm][pos];
      b2.y = vS[head * 16 + lm][pos + 1];
      ctxacc = wmma4(a, b2, ctxacc);
    }
    float s = 0.0f;
#pragma unroll
    for (int p2 = 0; p2 < 16; ++p2) s += eS[ech][egrp * 16 + p2];
    esum += s;
    __syncthreads();
  }

#pragma unroll
  for (int r = 0; r < 8; ++r) cbuf[wv][(r + 8 * lh) * 16 + lm] = ctxacc[r];
  esbuf[egrp][ech] = esum;
  __syncthreads();

  float* rec = partials + (size_t)(blockIdx.y * gridDim.x + blockIdx.x) * RECSZ;
  if (tid < 32) {
    rec[tid] = kmaxS[tid];                    // block max
    float s = 0.0f;
    for (int w = 0; w < 8; ++w) s += esbuf[w][tid];
    rec[32 + tid] = s;                        // block esum
  }
  for (int e = tid; e < 512; e += 256) {
    const int hh = e >> 8, idx = e & 255;
    float s = 0.0f;
    for (int w = hh; w < 8; w += 2) s += cbuf[w][idx];
    rec[64 + e] = s;                          // block ctx
  }
}

// ====== Pass 1b: fixed-order cross-block combine with exact rescaling =======
__global__ void k_reduce(const float* __restrict__ partials,
                         float* __restrict__ ctxn) {
  const int nb = blockIdx.x;
  const int e = threadIdx.x;  // head*256 + kc*16 + vc
  const int hh = e >> 8;
  const int kc = (e >> 4) & 15;
  const int rowch = hh * 16 + kc;
  const float* base = partials + (size_t)nb * 512 * RECSZ;

  float M = -3.402823e38f;
  for (int r = 0; r < 512; ++r) M = fmaxf(M, base[(size_t)r * RECSZ + rowch]);
  float cs = 0.0f, es = 0.0f;
  for (int r = 0; r < 512; ++r) {
    const float* p = base + (size_t)r * RECSZ;
    const float f = expf(p[rowch] - M);
    cs += p[64 + e] * f;
    es += p[32 + rowch] * f;
  }
  ctxn[nb * 512 + e] = cs / es;
}

// --- Pass 2: q proj (WMMA) -> ch softmax -> att -> out proj (WMMA) + resid ---
__global__ void __launch_bounds__(256, 2)
k_out(const float* __restrict__ x,
      const float* __restrict__ Wq, const float* __restrict__ bq,
      const float* __restrict__ Wr, const float* __restrict__ br,
      const float* __restrict__ ctxn, float* __restrict__ out) {
  __shared__ float qS[32][130];
  __shared__ float aS[32][130];
  __shared__ float ctxS[512];

  const int tid = threadIdx.x;
  const int lane = tid & 31, wv = tid >> 5;
  const int lm = lane & 15, lh = lane >> 4;
  const int nb = blockIdx.y;
  const size_t s0 = (size_t)blockIdx.x * 128 + (size_t)wv * 16;
  const float* xb = x + (size_t)nb * CIN * SLEN;

#if __has_builtin(__builtin_amdgcn_global_load_async_to_lds_b128)
  if (tid < 128) {
    __builtin_amdgcn_global_load_async_to_lds_b128(
        (glob_v4i*)(ctxn + nb * 512 + tid * 4),
        (lds_v4i*)(&ctxS[tid * 4]), 0, 0);
  }
#if __has_builtin(__builtin_amdgcn_s_wait_asynccnt)
  __builtin_amdgcn_s_wait_asynccnt(0);
#else
  asm volatile("s_wait_asynccnt 0" ::: "memory");
#endif
#else
  for (int i = tid; i < 512; i += 256) ctxS[i] = ctxn[nb * 512 + i];
#endif

  v2f barr[16];
#pragma unroll
  for (int kk = 0; kk < 16; ++kk) {   // batched x loads
    const int row = kk * 4 + 2 * lh;
    barr[kk].x = xb[(size_t)row * SLEN + s0 + lm];
    barr[kk].y = xb[(size_t)(row + 1) * SLEN + s0 + lm];
  }
  v8f q0 = vzero8(), q1 = vzero8();
#pragma unroll
  for (int kk = 0; kk < 16; ++kk) {
    const int row = kk * 4 + 2 * lh;
    v2f a0, a1;
    const float* p;
    p = Wq + lm * CIN + row;        a0.x = p[0]; a0.y = p[1];
    p = Wq + (16 + lm) * CIN + row; a1.x = p[0]; a1.y = p[1];
    q0 = wmma4(a0, barr[kk], q0);
    q1 = wmma4(a1, barr[kk], q1);
  }
  const int col = wv * 16 + lm;
#pragma unroll
  for (int r = 0; r < 8; ++r) {
    const int c0 = r + 8 * lh, c1 = 16 + r + 8 * lh;
    qS[c0][col] = q0[r] + bq[c0];
    qS[c1][col] = q1[r] + bq[c1];
  }
  __syncthreads();

  {
    const int p = tid & 127;
    const int hh = tid >> 7;
    float qv[16];
    float mx = -3.402823e38f;
#pragma unroll
    for (int kc = 0; kc < 16; ++kc) {
      qv[kc] = qS[hh * 16 + kc][p];
      mx = fmaxf(mx, qv[kc]);
    }
    float sm = 0.0f;
#pragma unroll
    for (int kc = 0; kc < 16; ++kc) { qv[kc] = __expf(qv[kc] - mx); sm += qv[kc]; }
    const float inv = 1.0f / sm;
#pragma unroll
    for (int vc = 0; vc < 16; ++vc) {
      float a = 0.0f;
#pragma unroll
      for (int kc = 0; kc < 16; ++kc) a += ctxS[hh * 256 + kc * 16 + vc] * qv[kc];
      aS[hh * 16 + vc][p] = a * inv;
    }
  }
  __syncthreads();

  v2f batt[8];
#pragma unroll
  for (int kk = 0; kk < 8; ++kk) {
    const int row = kk * 4 + 2 * lh;
    batt[kk].x = aS[row][col];
    batt[kk].y = aS[row + 1][col];
  }
  v8f o0 = vzero8(), o1 = vzero8(), o2 = vzero8(), o3 = vzero8();
#pragma unroll
  for (int kk = 0; kk < 8; ++kk) {
    const int row = kk * 4 + 2 * lh;
    v2f a0, a1, a2, a3;
    const float* p;
    p = Wr + lm * VCH + row;        a0.x = p[0]; a0.y = p[1];
    p = Wr + (16 + lm) * VCH + row; a1.x = p[0]; a1.y = p[1];
    p = Wr + (32 + lm) * VCH + row; a2.x = p[0]; a2.y = p[1];
    p = Wr + (48 + lm) * VCH + row; a3.x = p[0]; a3.y = p[1];
    o0 = wmma4(a0, batt[kk], o0);
    o1 = wmma4(a1, batt[kk], o1);
    o2 = wmma4(a2, batt[kk], o2);
    o3 = wmma4(a3, batt[kk], o3);
  }
  float* ob = out + (size_t)nb * CIN * SLEN;
#pragma unroll
  for (int r = 0; r < 8; ++r) {
    const int m = r + 8 * lh;
    const size_t i0 = (size_t)m * SLEN + s0 + lm;
    ob[i0] = o0[r] + br[m] + xb[i0];
    const size_t i1 = (size_t)(m + 16) * SLEN + s0 + lm;
    ob[i1] = o1[r] + br[m + 16] + xb[i1];
    const size_t i2 = (size_t)(m + 32) * SLEN + s0 + lm;
    ob[i2] = o2[r] + br[m + 32] + xb[i2];
    const size_t i3 = (size_t)(m + 48) * SLEN + s0 + lm;
    ob[i3] = o3[r] + br[m + 48] + xb[i3];
  }
}

extern "C" void kernel_launch(void* const* d_in, const int* in_sizes, int n_in,
                              void* d_out, int out_size, void* d_ws, size_t ws_size,
                              hipStream_t stream) {
  (void)in_sizes; (void)n_in; (void)out_size; (void)ws_size;
  const float* x  = (const float*)d_in[0];
  const float* Wk = (const float*)d_in[1];
  const float* bk = (const float*)d_in[2];
  const float* Wq = (const float*)d_in[3];
  const float* bq = (const float*)d_in[4];
  const float* Wv = (const float*)d_in[5];
  const float* bv = (const float*)d_in[6];
  const float* Wr = (const float*)d_in[7];
  const float* br = (const float*)d_in[8];
  float* out = (float*)d_out;

  float* ctxn     = (float*)((char*)d_ws + 256);   // 1024 floats
  float* partials = (float*)((char*)d_ws + 4608);  // 1024*RECSZ floats

  hipLaunchKernelGGL(k_stats, dim3(SLEN / 512, NBATCH), dim3(256), 0, stream,
                     x, Wk, bk, Wv, bv, partials);
  hipLaunchKernelGGL(k_reduce, dim3(NBATCH), dim3(512), 0, stream,
                     partials, ctxn);
  hipLaunchKernelGGL(k_out, dim3(SLEN / 128, NBATCH), dim3(256), 0, stream,
                     x, Wq, bq, Wr, br, ctxn, out);
}